// QuantumHybridHead_65481071401196
// MI455X (gfx1250) — compile-verified
//
#include <hip/hip_runtime.h>
#include <hip/hip_bf16.h>

// CDNA5 / gfx1250: wave32, WMMA 16x16 shapes.
typedef __attribute__((ext_vector_type(2))) float v2f;
typedef __attribute__((ext_vector_type(8))) float v8f;

#define N_QUBITS 10
#define D_IN     512
#define ROWS_PER_WAVE 16
#define WAVES_PER_BLK 8
#define ROWS_PER_BLK  (ROWS_PER_WAVE * WAVES_PER_BLK)   // 128

// out[b] = sigmoid( (1/10) * sum_w cos( (x@W + b)[b,w] + theta[w] ) )
// GEMM tile per wave: A = x[16 x 512] (f32), B = Wpad[512 x 16] (cols 10..15 zero),
// accumulated with V_WMMA_F32_16X16X4_F32 over 128 K-steps of 4.
__global__ __launch_bounds__(256, 2)
void qhead_wmma_kernel(const float* __restrict__ x,
                       const float* __restrict__ W,
                       const float* __restrict__ bvec,
                       const float* __restrict__ theta,
                       float* __restrict__ out)
{
    // W staged as K/2 x 16 float2 pairs: Wp[(k>>1)*16 + n] = { W[k][n], W[k+1][n] }
    // This matches the B-fragment layout of WMMA_F32_16X16X4_F32:
    //   lanes 0-15 need rows k0+0,k0+1 at col n; lanes 16-31 need rows k0+2,k0+3.
    __shared__ v2f   Wp[(D_IN / 2) * 16];          // 32 KB
    __shared__ float epi[WAVES_PER_BLK][16][17];   // accumulator spill, +1 pad vs banks
    __shared__ float bt[16];                       // b[n] + theta[n], padded

    const int tid = threadIdx.x;

    if (tid < 16)
        bt[tid] = (tid < N_QUBITS) ? (bvec[tid] + theta[tid]) : 0.0f;

    // Stage + pad W (512x10 -> pairs over K, 16 cols). 4096 float2 entries / 256 thr.
    for (int idx = tid; idx < (D_IN / 2) * 16; idx += 256) {
        const int kk = idx >> 4;      // K-pair index (k = 2*kk)
        const int n  = idx & 15;
        v2f v; v.x = 0.0f; v.y = 0.0f;
        if (n < N_QUBITS) {
            v.x = W[(2 * kk)     * N_QUBITS + n];
            v.y = W[(2 * kk + 1) * N_QUBITS + n];
        }
        Wp[idx] = v;
    }
    __syncthreads();

    const int lane = tid & 31;
    const int wave = tid >> 5;
    const int half = lane >> 4;       // 0: lanes 0-15 (K=0,1), 1: lanes 16-31 (K=2,3)
    const int ln   = lane & 15;       // M index for A, N index for B/C/D
    const int m0   = blockIdx.x * ROWS_PER_BLK + wave * ROWS_PER_WAVE;

    // A-fragment source: lane (half*16 + ln) holds x[m0+ln][k0 + 2*half + {0,1}]
    const float* xrow = x + (size_t)(m0 + ln) * D_IN + 2 * half;

    v8f acc = (v8f){0.f, 0.f, 0.f, 0.f, 0.f, 0.f, 0.f, 0.f};

#pragma unroll 8
    for (int kk = 0; kk < D_IN / 4; ++kk) {        // k0 = 4*kk
        v2f a  = *(const v2f*)(xrow + 4 * kk);                 // global_load_b64
        v2f bf = Wp[(2 * kk + half) * 16 + ln];                // ds_load_b64
        // D = A(16x4) * B(4x16) + C ; chained f32 accumulation
        acc = __builtin_amdgcn_wmma_f32_16x16x4_f32(
                  /*neg_a=*/false, a, /*neg_b=*/false, bf,
                  /*c_mod=*/(short)0, acc, /*reuse_a=*/false, /*reuse_b=*/false);
    }

    // C/D layout: VGPR r -> M = r + 8*half, N = ln. Spill tile to LDS.
#pragma unroll
    for (int r = 0; r < 8; ++r)
        epi[wave][r + 8 * half][ln] = acc[r];

    __syncthreads();   // also orders the DS writes above vs. reads below

    // One lane per output row: angles[m][n] = acc + b[n]; out = sigmoid(mean cos(angle+theta))
    if (lane < 16) {
        float s = 0.0f;
#pragma unroll
        for (int n = 0; n < N_QUBITS; ++n)
            s += __cosf(epi[wave][lane][n] + bt[n]);
        const float ev = s * (1.0f / N_QUBITS);
        out[m0 + lane] = 1.0f / (1.0f + __expf(-ev));
    }
}

extern "C" void kernel_launch(void* const* d_in, const int* in_sizes, int n_in,
                              void* d_out, int out_size, void* d_ws, size_t ws_size,
                              hipStream_t stream) {
    const float* x     = (const float*)d_in[0];   // [B, 512]
    const float* W     = (const float*)d_in[1];   // [512, 10]
    const float* bvec  = (const float*)d_in[2];   // [10]
    const float* theta = (const float*)d_in[3];   // [10]
    float*       out   = (float*)d_out;           // [B]

    const int B = in_sizes[0] / D_IN;             // 32768
    dim3 grid(B / ROWS_PER_BLK);                  // 256 blocks
    dim3 block(256);                              // 8 waves (wave32)
    qhead_wmma_kernel<<<grid, block, 0, stream>>>(x, W, bvec, theta, out);
}